// AnchorGNN_18433999634946
// MI455X (gfx1250) — compile-verified
//
#include <hip/hip_runtime.h>

typedef __attribute__((ext_vector_type(2))) float v2f;
typedef __attribute__((ext_vector_type(8))) float v8f;

#define GNN_N_NODES   100000
#define GNN_IN_DIM    48
#define GNN_HID_DIM   32
#define GNN_OUT_DIM   16
#define GNN_N_EDGES   1600000

// ---------------------------------------------------------------------------
// Native f32 atomic add (relaxed, device scope) -> global_atomic_add_f32
// ---------------------------------------------------------------------------
__device__ __forceinline__ void atom_add_f32(float* p, float v) {
    __hip_atomic_fetch_add(p, v, __ATOMIC_RELAXED, __HIP_MEMORY_SCOPE_AGENT);
}

// ---------------------------------------------------------------------------
// Zero a region with float4 stores
// ---------------------------------------------------------------------------
__global__ void gnn_zero_f4(float4* __restrict__ p, long long n4) {
    long long i = (long long)blockIdx.x * blockDim.x + threadIdx.x;
    if (i < n4) p[i] = make_float4(0.f, 0.f, 0.f, 0.f);
}

// ---------------------------------------------------------------------------
// Edge aggregation: out[row[e]] += feat[col[e]]   (DIM floats per edge)
// DIM/4 threads cooperate per edge; consecutive threads read consecutive
// 16B chunks of the same source row -> coalesced 192B/128B gathers.
// ---------------------------------------------------------------------------
template <int DIM>
__global__ void gnn_scatter_add(const float* __restrict__ feat,
                                const long long* __restrict__ row,
                                const long long* __restrict__ col,
                                float* __restrict__ out) {
    constexpr int CH = DIM / 4;                      // float4 chunks per row
    const long long total = (long long)GNN_N_EDGES * CH;
    long long tid = (long long)blockIdx.x * blockDim.x + threadIdx.x;
    if (tid >= total) return;
    int e = (int)(tid / CH);
    int c = (int)(tid % CH);
    long long src = col[e];
    long long dst = row[e];
    float4 v = reinterpret_cast<const float4*>(feat + src * DIM)[c];
    float* o = out + dst * DIM + c * 4;
    atom_add_f32(o + 0, v.x);
    atom_add_f32(o + 1, v.y);
    atom_add_f32(o + 2, v.z);
    atom_add_f32(o + 3, v.w);
}

// ---------------------------------------------------------------------------
// h = relu(agg @ W1^T + b1) via V_WMMA_F32_16X16X4_F32.
// One wave computes a 16-node x 32-col tile (2 WMMA accumulators), K=48 in
// steps of 4. Fragment layouts per CDNA5 ISA 7.12.2:
//   A 16x4 : lane m = lane%16; VGPR0 holds K={0 | 2}, VGPR1 K={1 | 3} by half
//   B 4x16 : lane n = lane%16; same K split by half
//   D 16x16: VGPR r -> M = r + 8*half, N = lane%16
// ---------------------------------------------------------------------------
__global__ void gnn_dense1_wmma(const float* __restrict__ agg,
                                const float* __restrict__ w1,   // [32][48]
                                const float* __restrict__ b1,   // [32]
                                float* __restrict__ h) {        // [N][32]
    int wid  = blockIdx.x * (blockDim.x >> 5) + (threadIdx.x >> 5);
    int lane = threadIdx.x & 31;
    int m0   = wid << 4;
    if (m0 >= GNN_N_NODES) return;          // wave-uniform: EXEC stays all-1s
    int half = lane >> 4;
    int ln   = lane & 15;

    const float* arow  = agg + (long long)(m0 + ln) * GNN_IN_DIM;
    const float* brow0 = w1 + (long long)(ln)      * GNN_IN_DIM; // n = ln
    const float* brow1 = w1 + (long long)(16 + ln) * GNN_IN_DIM; // n = 16+ln

    v8f acc0 = {};
    v8f acc1 = {};
#pragma unroll
    for (int k = 0; k < GNN_IN_DIM; k += 4) {
        int ko = k + 2 * half;
        float2 af = *reinterpret_cast<const float2*>(arow + ko);
        float2 b0 = *reinterpret_cast<const float2*>(brow0 + ko);
        float2 b1v = *reinterpret_cast<const float2*>(brow1 + ko);
        v2f a;  a.x = af.x;  a.y = af.y;
        v2f q0; q0.x = b0.x; q0.y = b0.y;
        v2f q1; q1.x = b1v.x; q1.y = b1v.y;
        acc0 = __builtin_amdgcn_wmma_f32_16x16x4_f32(false, a, false, q0,
                                                     (short)0, acc0, false, false);
        acc1 = __builtin_amdgcn_wmma_f32_16x16x4_f32(false, a, false, q1,
                                                     (short)0, acc1, false, false);
    }

    float bias0 = b1[ln];
    float bias1 = b1[16 + ln];
#pragma unroll
    for (int r = 0; r < 8; ++r) {
        int m = m0 + r + 8 * half;
        float v0 = acc0[r] + bias0; v0 = v0 > 0.f ? v0 : 0.f;
        float v1 = acc1[r] + bias1; v1 = v1 > 0.f ? v1 : 0.f;
        h[(long long)m * GNN_HID_DIM + ln]      = v0;
        h[(long long)m * GNN_HID_DIM + 16 + ln] = v1;
    }
}

// ---------------------------------------------------------------------------
// out = agg2 @ W2^T + b2  (16 output cols -> one WMMA accumulator, K=32)
// ---------------------------------------------------------------------------
__global__ void gnn_dense2_wmma(const float* __restrict__ agg2,
                                const float* __restrict__ w2,   // [16][32]
                                const float* __restrict__ b2,   // [16]
                                float* __restrict__ out) {      // [N][16]
    int wid  = blockIdx.x * (blockDim.x >> 5) + (threadIdx.x >> 5);
    int lane = threadIdx.x & 31;
    int m0   = wid << 4;
    if (m0 >= GNN_N_NODES) return;
    int half = lane >> 4;
    int ln   = lane & 15;

    const float* arow = agg2 + (long long)(m0 + ln) * GNN_HID_DIM;
    const float* brow = w2 + (long long)ln * GNN_HID_DIM;

    v8f acc = {};
#pragma unroll
    for (int k = 0; k < GNN_HID_DIM; k += 4) {
        int ko = k + 2 * half;
        float2 af = *reinterpret_cast<const float2*>(arow + ko);
        float2 bf = *reinterpret_cast<const float2*>(brow + ko);
        v2f a; a.x = af.x; a.y = af.y;
        v2f b; b.x = bf.x; b.y = bf.y;
        acc = __builtin_amdgcn_wmma_f32_16x16x4_f32(false, a, false, b,
                                                    (short)0, acc, false, false);
    }

    float bias = b2[ln];
#pragma unroll
    for (int r = 0; r < 8; ++r) {
        int m = m0 + r + 8 * half;
        out[(long long)m * GNN_OUT_DIM + ln] = acc[r] + bias;
    }
}

// ---------------------------------------------------------------------------
// Launcher
// ---------------------------------------------------------------------------
extern "C" void kernel_launch(void* const* d_in, const int* in_sizes, int n_in,
                              void* d_out, int out_size, void* d_ws, size_t ws_size,
                              hipStream_t stream) {
    const float*     x    = (const float*)d_in[0];
    const long long* eidx = (const long long*)d_in[1];   // int64 [2, N_EDGES]
    const float*     w1   = (const float*)d_in[2];
    const float*     b1   = (const float*)d_in[3];
    const float*     w2   = (const float*)d_in[4];
    const float*     b2   = (const float*)d_in[5];
    float*           out  = (float*)d_out;

    const long long* row = eidx;                 // edge_index[0]
    const long long* col = eidx + GNN_N_EDGES;   // edge_index[1]

    // Workspace layout: [agg (N*48) | agg2 (N*32) | h (N*32)]  (floats)
    float* agg  = (float*)d_ws;
    float* agg2 = agg  + (long long)GNN_N_NODES * GNN_IN_DIM;
    float* h    = agg2 + (long long)GNN_N_NODES * GNN_HID_DIM;

    // 1) zero both accumulators (contiguous region) : (48+32)*N floats
    {
        long long n4 = ((long long)GNN_N_NODES * (GNN_IN_DIM + GNN_HID_DIM)) / 4;
        int blocks = (int)((n4 + 255) / 256);
        gnn_zero_f4<<<blocks, 256, 0, stream>>>((float4*)d_ws, n4);
    }
    // 2) agg[row] += x[col]   (12 threads per edge)
    {
        long long total = (long long)GNN_N_EDGES * (GNN_IN_DIM / 4);
        int blocks = (int)((total + 255) / 256);
        gnn_scatter_add<GNN_IN_DIM><<<blocks, 256, 0, stream>>>(x, row, col, agg);
    }
    // 3) h = relu(agg @ W1^T + b1)   (6250 wave-tiles, 8 waves/block)
    {
        int tiles  = GNN_N_NODES / 16;           // 6250
        int blocks = (tiles + 7) / 8;
        gnn_dense1_wmma<<<blocks, 256, 0, stream>>>(agg, w1, b1, h);
    }
    // 4) agg2[row] += h[col]  (8 threads per edge)
    {
        long long total = (long long)GNN_N_EDGES * (GNN_HID_DIM / 4);
        int blocks = (int)((total + 255) / 256);
        gnn_scatter_add<GNN_HID_DIM><<<blocks, 256, 0, stream>>>(h, row, col, agg2);
    }
    // 5) out = agg2 @ W2^T + b2
    {
        int tiles  = GNN_N_NODES / 16;
        int blocks = (tiles + 7) / 8;
        gnn_dense2_wmma<<<blocks, 256, 0, stream>>>(agg2, w2, b2, out);
    }
}